// L12Solver_88399016886713
// MI455X (gfx1250) — compile-verified
//
#include <hip/hip_runtime.h>
#include <hip/hip_bf16.h>

// Problem constants (match reference)
#define BSZ   256
#define MDIM  128
#define NDIM  1024
#define NITER 100
#define ALPHA 0.1f

typedef float v2f __attribute__((ext_vector_type(2)));
typedef float v8f __attribute__((ext_vector_type(8)));

// ---------------------------------------------------------------------------
// Kernel 1: AAT[b] = A[b] * A[b]^T  (128x128 = 128x1024 * 1024x128), fp32 WMMA
// One block per batch, 8 waves. Wave w computes rows [16w, 16w+16) x all 128
// cols as 8 accumulated 16x16 tiles via V_WMMA_F32_16X16X4_F32.
//
// f32 16x4 A-fragment layout (ISA 7.12.2): lane l: m = l&15, kpair = (l>>4)*2,
// holds K=kpair, kpair+1 in its two VGPRs. The 4x16 B-fragment (row striped
// across lanes per VGPR) for B = (A rows rn..rn+15)^T has the *same* lane
// addressing with the row base swapped.
// ---------------------------------------------------------------------------
__global__ __launch_bounds__(256) void aat_kernel(const float* __restrict__ A,
                                                  float* __restrict__ AAT) {
    const int batch = blockIdx.x;
    const float* __restrict__ Ab = A + (size_t)batch * MDIM * NDIM;
    float* __restrict__ Cb = AAT + (size_t)batch * MDIM * MDIM;

    const int wave = threadIdx.x >> 5;   // 0..7 -> row block
    const int lane = threadIdx.x & 31;
    const int q    = lane & 15;          // row within 16-row strip
    const int kb   = (lane >> 4) << 1;   // 0 or 2 (k sub-offset)
    const int rm   = wave << 4;

    v8f acc[8];
    const v8f vzero = {0.f, 0.f, 0.f, 0.f, 0.f, 0.f, 0.f, 0.f};
#pragma unroll
    for (int nt = 0; nt < 8; ++nt) acc[nt] = vzero;

    const float* arow = Ab + (size_t)(rm + q) * NDIM + kb;
    const float* brow = Ab + (size_t)q * NDIM + kb;

    for (int k = 0; k < NDIM; k += 4) {
        v2f a = *(const v2f*)(arow + k);
#pragma unroll
        for (int nt = 0; nt < 8; ++nt) {
            v2f bfrag = *(const v2f*)(brow + (size_t)(nt << 4) * NDIM + k);
            acc[nt] = __builtin_amdgcn_wmma_f32_16x16x4_f32(
                false, a, false, bfrag, (short)0, acc[nt], false, false);
        }
    }

    // C/D layout: VGPR v, lanes 0-15: row v; lanes 16-31: row v+8; col = lane&15
    const int rowadd = (lane >> 4) << 3;
#pragma unroll
    for (int nt = 0; nt < 8; ++nt) {
#pragma unroll
        for (int v = 0; v < 8; ++v) {
            Cb[(size_t)(rm + v + rowadd) * MDIM + (nt << 4) + q] = acc[nt][v];
        }
    }
}

// ---------------------------------------------------------------------------
// Kernel 2: in-place Gauss-Jordan inverse of each 128x128 AAT (SPD, no
// pivoting needed). Column-parallel elimination: every LDS access is either
// lane-consecutive (conflict-free) or a broadcast. Exactly 64KB static LDS.
// ---------------------------------------------------------------------------
__global__ __launch_bounds__(256) void inv_kernel(float* __restrict__ G) {
    __shared__ float Msh[MDIM * MDIM];   // 64 KB
    const int tid = threadIdx.x;
    float* __restrict__ Gb = G + (size_t)blockIdx.x * MDIM * MDIM;

    for (int i = tid; i < MDIM * MDIM; i += 256) Msh[i] = Gb[i];

    for (int k = 0; k < MDIM; ++k) {
        __syncthreads();                       // prior writes visible
        const float pinv = 1.0f / Msh[k * MDIM + k];   // broadcast read
        __syncthreads();                       // everyone read pivot
        if (tid < MDIM) {                      // scale pivot row
            Msh[k * MDIM + tid] = (tid == k) ? pinv : Msh[k * MDIM + tid] * pinv;
        }
        __syncthreads();
        {   // eliminate: 2 threads per column, disjoint row halves
            const int j  = tid & 127;
            const int r0 = (tid >> 7) << 6;    // 0 or 64
            if (j != k) {
                const float pv = Msh[k * MDIM + j];
                for (int r = r0; r < r0 + 64; ++r) {
                    if (r != k) {
                        const float f = Msh[r * MDIM + k];   // broadcast
                        Msh[r * MDIM + j] -= f * pv;         // coalesced
                    }
                }
            }
        }
        __syncthreads();
        if (tid < MDIM && tid != k) {          // rewrite pivot column
            Msh[tid * MDIM + k] = -Msh[tid * MDIM + k] * pinv;
        }
    }
    __syncthreads();
    for (int i = tid; i < MDIM * MDIM; i += 256) Gb[i] = Msh[i];
}

// ---------------------------------------------------------------------------
// Kernel 3: 100 Douglas-Rachford iterations per batch, state in LDS.
// ---------------------------------------------------------------------------
__device__ __forceinline__ float wave_reduce(float v) {
#pragma unroll
    for (int off = 16; off > 0; off >>= 1) v += __shfl_down(v, off, 32);
    return v;
}

__global__ __launch_bounds__(256) void solve_kernel(const float* __restrict__ A,
                                                    const float* __restrict__ bvec,
                                                    const float* __restrict__ D1,
                                                    const float* __restrict__ D2,
                                                    const float* __restrict__ G,
                                                    float* __restrict__ out) {
    __shared__ float zs[NDIM], ys[NDIM], xh[NDIM], thr[NDIM], dinv[NDIM];
    __shared__ float ts[MDIM], us[MDIM], bsh[MDIM];

    const int batch = blockIdx.x;
    const int tid   = threadIdx.x;
    const int wave  = tid >> 5;
    const int lane  = tid & 31;
    const float* __restrict__ Ab = A + (size_t)batch * MDIM * NDIM;
    const float* __restrict__ Gb = G + (size_t)batch * MDIM * MDIM;

    for (int n = tid; n < NDIM; n += 256) {
        const float d1 = D1[(size_t)batch * NDIM + n];
        const float d2 = D2[(size_t)batch * NDIM + n];
        thr[n]  = ALPHA * fabsf(d1);
        dinv[n] = 1.0f / (1.0f + 2.0f * ALPHA * d2 * d2);
        zs[n]   = 0.0f;
    }
    if (tid < MDIM) bsh[tid] = bvec[(size_t)batch * MDIM + tid];
    __syncthreads();

    for (int iter = 0; iter < NITER; ++iter) {
        // Stage 1: soft-threshold, y = 2*x_half - z
        for (int n = tid; n < NDIM; n += 256) {
            const float z = zs[n];
            const float t = thr[n];
            float x;
            if (z - t > 0.0f)       x = (z - t) * dinv[n];
            else if (z + t < 0.0f)  x = (z + t) * dinv[n];
            else                    x = 0.0f;
            xh[n] = x;
            ys[n] = 2.0f * x - z;
        }
        __syncthreads();

        // Stage 2: t = A*y - b   (wave per row, coalesced lane-strided reads)
#pragma unroll 1
        for (int mi = 0; mi < 16; ++mi) {
            const int m = (wave << 4) + mi;
            const float* __restrict__ row = Ab + (size_t)m * NDIM;
            float s = 0.0f;
#pragma unroll 4
            for (int j = lane; j < NDIM; j += 32) s += row[j] * ys[j];
            s = wave_reduce(s);
            if (lane == 0) ts[m] = s - bsh[m];
        }
        __syncthreads();

        // Stage 3: u = G * t  (G streamed from L2, coalesced)
#pragma unroll 1
        for (int mi = 0; mi < 16; ++mi) {
            const int m = (wave << 4) + mi;
            const float* __restrict__ grow = Gb + (size_t)m * MDIM;
            float s = 0.0f;
#pragma unroll 4
            for (int j = lane; j < MDIM; j += 32) s += grow[j] * ts[j];
            s = wave_reduce(s);
            if (lane == 0) us[m] = s;
        }
        __syncthreads();

        // Stage 4: corr = A^T*u; x_new = y - corr; z += x_new - x_half
        for (int n = tid; n < NDIM; n += 256) {
            float c = 0.0f;
#pragma unroll 4
            for (int m = 0; m < MDIM; ++m) c += Ab[(size_t)m * NDIM + n] * us[m];
            const float xn = ys[n] - c;
            zs[n] = zs[n] + xn - xh[n];
            if (iter == NITER - 1) out[(size_t)batch * NDIM + n] = xn;
        }
        __syncthreads();
    }
}

// ---------------------------------------------------------------------------
extern "C" void kernel_launch(void* const* d_in, const int* in_sizes, int n_in,
                              void* d_out, int out_size, void* d_ws, size_t ws_size,
                              hipStream_t stream) {
    const float* A  = (const float*)d_in[0];   // (256,128,1024)
    const float* b  = (const float*)d_in[1];   // (256,128)
    const float* D1 = (const float*)d_in[2];   // (256,1024)
    const float* D2 = (const float*)d_in[3];   // (256,1024)
    float* out      = (float*)d_out;           // (256,1024)

    float* AAT = (float*)d_ws;                 // 256*128*128*4 = 16 MB scratch

    aat_kernel  <<<BSZ, 256, 0, stream>>>(A, AAT);
    inv_kernel  <<<BSZ, 256, 0, stream>>>(AAT);
    solve_kernel<<<BSZ, 256, 0, stream>>>(A, b, D1, D2, AAT, out);
}